// RoutingFunction_18442589569222
// MI455X (gfx1250) — compile-verified
//
#include <hip/hip_runtime.h>
#include <hip/hip_bf16.h>
#include <math.h>

// Problem constants (from reference)
#define NB 256
#define NDIM 512
#define NHW 1024
#define NFREQ 256
#define NE 16
#define TAU_F 1.0f
#define NOISE_STD_F (1.0f / 16.0f)
#define INV_SQRT2_F 0.7071067811865476f

typedef __attribute__((ext_vector_type(2))) float v2f;
typedef __attribute__((ext_vector_type(4))) float v4f;
typedef __attribute__((ext_vector_type(8))) float v8f;

// ---------------------------------------------------------------------------
// Kernel 1: pooled[b][d] = mean over H*W of x[b][d][:][:]
// One wave32 per (b,d) pair; 4KB contiguous per pair; pure b128 streaming.
// x is 512 MB read exactly once (> 192 MB L2, zero reuse) -> non-temporal
// loads so the stream doesn't thrash L2. Block = 256 threads = 8 waves.
// ---------------------------------------------------------------------------
__global__ void pool_mean_kernel(const float* __restrict__ x,
                                 float* __restrict__ pooled) {
    const int wave = threadIdx.x >> 5;
    const int lane = threadIdx.x & 31;
    const long long pair = (long long)blockIdx.x * 8 + wave;   // < NB*NDIM
    const v4f* __restrict__ src = (const v4f*)(x + pair * (long long)NHW);

    float4 acc = make_float4(0.f, 0.f, 0.f, 0.f);
#pragma unroll
    for (int i = 0; i < 8; ++i) {
        v4f v = __builtin_nontemporal_load(&src[lane + i * 32]); // b128 NT
        acc.x += v[0]; acc.y += v[1]; acc.z += v[2]; acc.w += v[3];
    }
    float s = (acc.x + acc.y) + (acc.z + acc.w);
    // wave32 butterfly reduction
#pragma unroll
    for (int off = 16; off > 0; off >>= 1)
        s += __shfl_xor(s, off, 32);
    if (lane == 0)
        pooled[pair] = s * (1.0f / (float)NHW);   // RT store: reused by kernel 2
}

// ---------------------------------------------------------------------------
// Kernel 2: logits via V_WMMA_F32_16X16X4_F32, then softmax/top-2/losses.
// One block of 512 threads = 16 waves; wave w computes logits rows
// [16w, 16w+16) x 16 experts as one 16x16 WMMA C tile.
// ---------------------------------------------------------------------------
__global__ void
__launch_bounds__(512)
router_kernel(const float* __restrict__ pooled,
              const float* __restrict__ freq_emb,
              const float* __restrict__ w_gate,     // (NE, NDIM)
              const float* __restrict__ w_freq,     // (NE, NFREQ)
              const float* __restrict__ noise,      // (NB, NE)
              const float* __restrict__ complexity, // (NE)
              float* __restrict__ out) {
    __shared__ float logits_s[NB][NE];
    __shared__ float imp_sum[NE];
    __shared__ float p_sum[NE];

    const int tid  = threadIdx.x;        // 0..511
    const int wave = tid >> 5;           // 0..15 -> row tile
    const int lane = tid & 31;
    const int n    = lane & 15;          // N (expert) index, also M for A rows
    const int kh   = lane >> 4;          // K half-select (0 -> K0/K1, 1 -> K2/K3)
    const int row0 = wave * 16;

    // ---- WMMA accumulation: D = pooled @ w_gate^T + freq @ w_freq^T ----
    v8f c = {};
    for (int kb = 0; kb < NDIM; kb += 4) {
        v2f a, b;
        const float* ap = pooled + (row0 + n) * NDIM + kb + kh * 2;
        a[0] = ap[0]; a[1] = ap[1];
        const float* bp = w_gate + n * NDIM + kb + kh * 2;  // B[k][n] = w_gate[n][k]
        b[0] = bp[0]; b[1] = bp[1];
        c = __builtin_amdgcn_wmma_f32_16x16x4_f32(
            /*neg_a=*/false, a, /*neg_b=*/false, b,
            /*c_mod=*/(short)0, c, /*reuse_a=*/false, /*reuse_b=*/false);
    }
    for (int kb = 0; kb < NFREQ; kb += 4) {
        v2f a, b;
        const float* ap = freq_emb + (row0 + n) * NFREQ + kb + kh * 2;
        a[0] = ap[0]; a[1] = ap[1];
        const float* bp = w_freq + n * NFREQ + kb + kh * 2;
        b[0] = bp[0]; b[1] = bp[1];
        c = __builtin_amdgcn_wmma_f32_16x16x4_f32(
            false, a, false, b, (short)0, c, false, false);
    }
    // C/D layout: lane L, VGPR r -> M = r + 8*(L>=16), N = L&15
#pragma unroll
    for (int r = 0; r < 8; ++r)
        logits_s[row0 + r + 8 * kh][n] = c[r];

    if (tid < NE) { imp_sum[tid] = 0.f; p_sum[tid] = 0.f; }
    __syncthreads();

    // ---- per-row gating math (256 rows, one thread each) ----
    if (tid < NB) {
        float lg[NE], nz[NE];
        float maxl = -3.4e38f, maxn = -3.4e38f;
#pragma unroll
        for (int e = 0; e < NE; ++e) {
            lg[e] = logits_s[tid][e];
            nz[e] = lg[e] + noise[tid * NE + e] * NOISE_STD_F;
            maxl = fmaxf(maxl, lg[e]);
            maxn = fmaxf(maxn, nz[e]);
        }
        float suml = 0.f, sumn = 0.f;
#pragma unroll
        for (int e = 0; e < NE; ++e) {
            suml += __expf(lg[e] - maxl);
            sumn += __expf(nz[e] - maxn);
        }
        const float rsl = 1.0f / suml;
        const float rsn = 1.0f / sumn;

        // top-2 of noisy logits (== top-2 of softmax(noisy); ties -> lower idx)
        int i1 = 0;
#pragma unroll
        for (int e = 1; e < NE; ++e) if (nz[e] > nz[i1]) i1 = e;
        int i2 = (i1 == 0) ? 1 : 0;
#pragma unroll
        for (int e = 0; e < NE; ++e)
            if (e != i1 && nz[e] > nz[i2]) i2 = e;

        const float g1 = __expf(nz[i1] - maxn) * rsn;
        const float g2 = __expf(nz[i2] - maxn) * rsn;

        // gates (dense 256x16), indices, values
#pragma unroll
        for (int e = 0; e < NE; ++e)
            out[tid * NE + e] = (e == i1) ? g1 : ((e == i2) ? g2 : 0.f);
        out[NB * NE + tid * 2 + 0]          = (float)i1;
        out[NB * NE + tid * 2 + 1]          = (float)i2;
        out[NB * NE + NB * 2 + tid * 2 + 0] = g1;
        out[NB * NE + NB * 2 + tid * 2 + 1] = g2;

        // loss accumulators: clean softmax + load probability p
        const float thr = nz[i2];   // K-th (2nd) largest noisy logit
#pragma unroll
        for (int e = 0; e < NE; ++e) {
            const float sm = __expf(lg[e] - maxl) * rsl;
            const float z  = (thr - lg[e]) * (1.0f / NOISE_STD_F);
            const float p  = 0.5f * erfcf(z * INV_SQRT2_F);
            atomicAdd(&imp_sum[e], sm);
            atomicAdd(&p_sum[e], p);
        }
    }
    __syncthreads();

    // ---- final scalar reductions (ddof=1 std) ----
    if (tid == 0) {
        float imp[NE], pm[NE];
        float mi = 0.f, mp = 0.f;
#pragma unroll
        for (int e = 0; e < NE; ++e) {
            imp[e] = imp_sum[e] * (complexity[e] * TAU_F);
            pm[e]  = p_sum[e] * (1.0f / (float)NB);
            mi += imp[e];
            mp += pm[e];
        }
        mi *= (1.0f / (float)NE);
        mp *= (1.0f / (float)NE);
        float vi = 0.f, vp = 0.f;
#pragma unroll
        for (int e = 0; e < NE; ++e) {
            const float di = imp[e] - mi;
            const float dp = pm[e] - mp;
            vi += di * di;
            vp += dp * dp;
        }
        vi *= (1.0f / (float)(NE - 1));   // unbiased variance
        vp *= (1.0f / (float)(NE - 1));
        const float ri = 1.0f / (mi + 1e-8f);
        const float rp = 1.0f / (mp + 1e-8f);
        const float loss_imp  = vi * ri * ri;   // (std/mean)^2 == var/mean^2
        const float loss_load = vp * rp * rp;
        out[NB * NE + NB * 4] = 0.5f * loss_imp + 0.5f * loss_load;
    }
}

// ---------------------------------------------------------------------------
extern "C" void kernel_launch(void* const* d_in, const int* in_sizes, int n_in,
                              void* d_out, int out_size, void* d_ws, size_t ws_size,
                              hipStream_t stream) {
    const float* x          = (const float*)d_in[0];
    const float* freq_emb   = (const float*)d_in[1];
    const float* w_gate     = (const float*)d_in[2];
    const float* w_freq     = (const float*)d_in[3];
    const float* noise      = (const float*)d_in[4];
    const float* complexity = (const float*)d_in[5];
    float* out = (float*)d_out;
    float* pooled = (float*)d_ws;   // NB*NDIM floats = 512 KB scratch

    // Kernel 1: 131072 (b,d) pairs, 8 waves/block -> 16384 blocks
    pool_mean_kernel<<<(NB * NDIM) / 8, 256, 0, stream>>>(x, pooled);

    // Kernel 2: single workgroup, 16 waves, WMMA GEMM + gating math
    router_kernel<<<1, 512, 0, stream>>>(pooled, freq_emb, w_gate, w_freq,
                                         noise, complexity, out);
}